// GIN_noparam_10273561772522
// MI455X (gfx1250) — compile-verified
//
#include <hip/hip_runtime.h>

// GIN (eps = -1) => two rounds of in-neighbor mean aggregation + concat.
// CSR built per call (int atomics + LDS scans), then gather-based
// aggregation: 1 wave32 per node, 128 floats as 32 lanes x float4
// (global_load_b128), cooperative neighbor-index fetch + __shfl broadcast,
// global_prefetch_b8 on the next neighbor row. No float atomics in hot loops.
// Feature copy exercises the CDNA5 async-to-LDS engine (ASYNCcnt) when the
// builtins are available.

#define N_FEAT 128
#define OUT_STRIDE 384

#if defined(__AMDGCN__) && \
    __has_builtin(__builtin_amdgcn_global_load_async_to_lds_b128) && \
    __has_builtin(__builtin_amdgcn_global_store_async_from_lds_b128) && \
    __has_builtin(__builtin_amdgcn_s_wait_asynccnt)
#define HAVE_ASYNC_LDS 1
typedef int v4i __attribute__((vector_size(16)));
typedef __attribute__((address_space(1))) v4i GV4i;  // global
typedef __attribute__((address_space(3))) v4i LV4i;  // LDS
#else
#define HAVE_ASYNC_LDS 0
#endif

__global__ void zero_kernel(int* __restrict__ p, int n) {
  int i = blockIdx.x * blockDim.x + threadIdx.x;
  if (i < n) p[i] = 0;
}

// out[:, 0:128] = feature  (streamed through LDS via the async engine)
__global__ void copy_feat_kernel(const float* __restrict__ feat,
                                 float* __restrict__ out, int n) {
  int t = blockIdx.x * blockDim.x + threadIdx.x;
  int node = t >> 5;
  int lane = t & 31;
  if (node >= n) return;
#if HAVE_ASYNC_LDS
  __shared__ float stage[8][N_FEAT];  // 8 waves/block * 512B
  int w = (threadIdx.x >> 5) & 7;
  float* lgen = &stage[w][lane * 4];
  __builtin_amdgcn_global_load_async_to_lds_b128(
      (GV4i*)(feat + (size_t)node * N_FEAT + lane * 4), (LV4i*)lgen, 0, 0);
  __builtin_amdgcn_s_wait_asynccnt(0);
  __builtin_amdgcn_global_store_async_from_lds_b128(
      (GV4i*)(out + (size_t)node * OUT_STRIDE + lane * 4), (LV4i*)lgen, 0, 0);
  __builtin_amdgcn_s_wait_asynccnt(0);
#else
  float4 v = ((const float4*)(feat + (size_t)node * N_FEAT))[lane];
  ((float4*)(out + (size_t)node * OUT_STRIDE))[lane] = v;
#endif
}

__global__ void count_kernel(const int* __restrict__ dst, int* __restrict__ deg, int e) {
  int i = blockIdx.x * blockDim.x + threadIdx.x;
  if (i < e) atomicAdd(&deg[dst[i]], 1);
}

// Per-1024-block inclusive->exclusive Hillis-Steele scan in LDS; also emits
// per-node deg_inv (1/deg, 0 for isolated nodes) and per-block totals.
__global__ void scanA_kernel(const int* __restrict__ deg, int* __restrict__ rowStart,
                             int* __restrict__ blockSums, float* __restrict__ deg_inv,
                             int n) {
  __shared__ int tmp[1024];
  int t = threadIdx.x;
  int i = blockIdx.x * 1024 + t;
  int v = (i < n) ? deg[i] : 0;
  if (i < n) deg_inv[i] = (v > 0) ? (1.0f / (float)v) : 0.0f;
  tmp[t] = v;
  __syncthreads();
  for (int off = 1; off < 1024; off <<= 1) {
    int x = (t >= off) ? tmp[t - off] : 0;
    __syncthreads();
    tmp[t] += x;
    __syncthreads();
  }
  if (i < n) rowStart[i] = tmp[t] - v;            // exclusive
  if (t == 1023) blockSums[blockIdx.x] = tmp[t];  // block total
}

__global__ void scanB_kernel(int* __restrict__ blockSums, int nb) {
  __shared__ int tmp[1024];
  int t = threadIdx.x;
  int v = (t < nb) ? blockSums[t] : 0;
  tmp[t] = v;
  __syncthreads();
  for (int off = 1; off < 1024; off <<= 1) {
    int x = (t >= off) ? tmp[t - off] : 0;
    __syncthreads();
    tmp[t] += x;
    __syncthreads();
  }
  if (t < nb) blockSums[t] = tmp[t] - v;  // exclusive block offsets
}

__global__ void scanC_kernel(int* __restrict__ rowStart,
                             const int* __restrict__ blockSums, int n) {
  int i = blockIdx.x * 1024 + threadIdx.x;
  if (i < n) rowStart[i] += blockSums[blockIdx.x];
}

// Bucket-fill CSR adjacency: adj[rowStart[dst] + slot] = src
__global__ void fill_kernel(const int* __restrict__ src, const int* __restrict__ dst,
                            const int* __restrict__ rowStart, int* __restrict__ cursor,
                            int* __restrict__ adj, int e) {
  int i = blockIdx.x * blockDim.x + threadIdx.x;
  if (i >= e) return;
  int d = dst[i];
  int pos = atomicAdd(&cursor[d], 1);
  adj[rowStart[d] + pos] = src[i];
}

// One wave32 per node; each lane owns 4 of the 128 features (float4 / b128).
// Neighbor indices fetched cooperatively (one coalesced b32 per 32 neighbors),
// broadcast via __shfl; global_prefetch_b8 pipelines the next row out of L2.
__global__ void agg_kernel(const float* __restrict__ in, int inStride,
                           float* __restrict__ out, int outStride,
                           const int* __restrict__ rowStart,
                           const int* __restrict__ deg,
                           const float* __restrict__ deg_inv,
                           const int* __restrict__ adj, int n) {
  int t = blockIdx.x * blockDim.x + threadIdx.x;
  int node = t >> 5;
  int lane = t & 31;
  if (node >= n) return;
  int beg = rowStart[node];
  int cnt = deg[node];
  float4 acc = make_float4(0.f, 0.f, 0.f, 0.f);
  for (int cbase = 0; cbase < cnt; cbase += 32) {
    int rem = cnt - cbase;
    int m = rem < 32 ? rem : 32;
    int idx = (lane < m) ? adj[beg + cbase + lane] : 0;  // coalesced chunk
    {
      int s0 = __shfl(idx, 0, 32);
      __builtin_prefetch(in + (size_t)s0 * inStride + lane * 4, 0, 3);
    }
    for (int j = 0; j < m; ++j) {
      int s = __shfl(idx, j, 32);
      if (j + 1 < m) {
        int s2 = __shfl(idx, j + 1, 32);
        __builtin_prefetch(in + (size_t)s2 * inStride + lane * 4, 0, 3);
      }
      float4 f = ((const float4*)(in + (size_t)s * inStride))[lane];
      acc.x += f.x; acc.y += f.y; acc.z += f.z; acc.w += f.w;
    }
  }
  float mval = deg_inv[node];
  acc.x *= mval; acc.y *= mval; acc.z *= mval; acc.w *= mval;
  ((float4*)(out + (size_t)node * outStride))[lane] = acc;
}

extern "C" void kernel_launch(void* const* d_in, const int* in_sizes, int n_in,
                              void* d_out, int out_size, void* d_ws, size_t ws_size,
                              hipStream_t stream) {
  (void)n_in; (void)out_size; (void)ws_size;

  const float* feat = (const float*)d_in[0];
  const int*   ei   = (const int*)d_in[1];
  const int N = in_sizes[0] / N_FEAT;  // 100000
  const int E = in_sizes[1] / 2;       // 1600000
  const int* src = ei;                 // edge_index[0]
  const int* dst = ei + E;             // edge_index[1]
  float* out = (float*)d_out;

  // Workspace layout (~8 MB total)
  int* deg       = (int*)d_ws;         // N
  int* cursor    = deg + N;            // N   (contiguous with deg -> one zero pass)
  int* rowStart  = cursor + N;         // N
  int* blockSums = rowStart + N;       // up to 1024
  int* adj       = blockSums + 1024;   // E
  float* deg_inv = (float*)(adj + E);  // N

  const int nbScan   = (N + 1023) / 1024;  // 98 for N=100000
  const int eblocks  = (E + 255) / 256;
  const int nwaves   = (int)(((size_t)N * 32 + 255) / 256);

  zero_kernel<<<(2 * N + 255) / 256, 256, 0, stream>>>(deg, 2 * N);
  count_kernel<<<eblocks, 256, 0, stream>>>(dst, deg, E);
  scanA_kernel<<<nbScan, 1024, 0, stream>>>(deg, rowStart, blockSums, deg_inv, N);
  scanB_kernel<<<1, 1024, 0, stream>>>(blockSums, nbScan);
  scanC_kernel<<<nbScan, 1024, 0, stream>>>(rowStart, blockSums, N);
  fill_kernel<<<eblocks, 256, 0, stream>>>(src, dst, rowStart, cursor, adj, E);

  copy_feat_kernel<<<nwaves, 256, 0, stream>>>(feat, out, N);
  // layer 1: mean over neighbors of feature -> out[:, 128:256]
  agg_kernel<<<nwaves, 256, 0, stream>>>(feat, N_FEAT, out + 128, OUT_STRIDE,
                                         rowStart, deg, deg_inv, adj, N);
  // layer 2: mean over neighbors of h1 (read back from out) -> out[:, 256:384]
  agg_kernel<<<nwaves, 256, 0, stream>>>(out + 128, OUT_STRIDE, out + 256, OUT_STRIDE,
                                         rowStart, deg, deg_inv, adj, N);
}